// ObjectSelect_37993280700614
// MI455X (gfx1250) — compile-verified
//
#include <hip/hip_runtime.h>
#include <math.h>
#include <stdint.h>

#define NEG_INF (-__builtin_huge_valf())

namespace {
constexpr int B_     = 8;
constexpr int N_     = 2000;
constexpr int C_     = 21;
constexpr int FG     = 20;     // foreground classes
constexpr int SORT_N = 2048;   // padded, power of two
constexpr float THRESH = 0.05f;
constexpr float NMS_T  = 0.3f;
constexpr int   MAXPI  = 50;
}

// ---------------------------------------------------------------------------
// Kernel 1: decode boxes for every (b, roi, fg-class); class-major layout.
// ---------------------------------------------------------------------------
__global__ __launch_bounds__(256) void decode_kernel(
    const float* __restrict__ rois, const float* __restrict__ deltas,
    const float* __restrict__ cls_prob, const float* __restrict__ im_info,
    float* __restrict__ decBoxes, float* __restrict__ decScores, int path)
{
    int id = blockIdx.x * 256 + threadIdx.x;
    if (id >= B_ * N_ * FG) return;
    int k = id % FG;
    int n = (id / FG) % N_;
    int b = id / (FG * N_);

    const float* r = rois + (size_t)(b * N_ + n) * 5;
    float x1 = r[1], y1 = r[2], x2 = r[3], y2 = r[4];
    float w  = x2 - x1 + 1.0f;
    float h  = y2 - y1 + 1.0f;
    float cx = x1 + 0.5f * w;
    float cy = y1 + 0.5f * h;

    int cls = k + 1;
    const float* d = deltas + (size_t)(b * N_ + n) * (4 * C_) + cls * 4;
    float dx = d[0] * 0.1f, dy = d[1] * 0.1f;
    float dw = d[2] * 0.2f, dh = d[3] * 0.2f;

    float pcx = dx * w + cx;
    float pcy = dy * h + cy;
    float pw  = expf(dw) * w;
    float ph  = expf(dh) * h;

    const float* info = im_info + b * 3;
    float W1 = info[1] - 1.0f;
    float H1 = info[0] - 1.0f;
    float sc = info[2];

    float bx1 = fminf(fmaxf(pcx - 0.5f * pw, 0.0f), W1) / sc;
    float by1 = fminf(fmaxf(pcy - 0.5f * ph, 0.0f), H1) / sc;
    float bx2 = fminf(fmaxf(pcx + 0.5f * pw, 0.0f), W1) / sc;
    float by2 = fminf(fmaxf(pcy + 0.5f * ph, 0.0f), H1) / sc;

    size_t base = ((size_t)((path * B_ + b) * FG + k)) * N_ + n;
    decBoxes[base * 4 + 0] = bx1;
    decBoxes[base * 4 + 1] = by1;
    decBoxes[base * 4 + 2] = bx2;
    decBoxes[base * 4 + 3] = by2;
    decScores[base] = cls_prob[(size_t)(b * N_ + n) * C_ + cls];
}

// ---------------------------------------------------------------------------
// Kernel 2: per (path, image, class): threshold, stable bitonic sort in LDS,
// async global->LDS gather of boxes (gfx1250 path), greedy NMS, compaction.
// ---------------------------------------------------------------------------
__global__ __launch_bounds__(256) void nms_kernel(
    const float* __restrict__ decBoxes, const float* __restrict__ decScores,
    float* __restrict__ keptScore, int* __restrict__ keptFlat,
    int* __restrict__ keptCount,
    float* __restrict__ cbScore, int* __restrict__ cbFlat,
    float* __restrict__ cbBox, float* __restrict__ fallback)
{
    __shared__ float sScore[SORT_N];
    __shared__ int   sIdx[SORT_N];
    __shared__ float sBox[SORT_N * 4];
    __shared__ unsigned char sSupp[SORT_N];
    __shared__ int sValid, sCnt;

    int tid  = threadIdx.x;
    int c    = blockIdx.x;
    int b    = blockIdx.y;
    int path = blockIdx.z;
    int pIdx = (path * B_ + b) * FG + c;

    const float* scBase  = decScores + (size_t)pIdx * N_;
    const float* boxBase = decBoxes  + (size_t)pIdx * N_ * 4;

    if (tid == 0) { sValid = 0; sCnt = 0; }
    __syncthreads();

    int myValid = 0;
    for (int t = tid; t < SORT_N; t += 256) {
        float s = NEG_INF;
        if (t < N_) {
            float v = scBase[t];
            if (v > THRESH) { s = v; myValid++; }
        }
        sScore[t] = s;
        sIdx[t]   = t;   // pad idx >= N_ -> sorts after every real -inf entry
    }
    if (myValid) atomicAdd(&sValid, myValid);

    // Bitonic sort: score descending, index ascending on ties (matches
    // jnp.argsort(-masked) stable order).
    for (int kk = 2; kk <= SORT_N; kk <<= 1) {
        for (int j = kk >> 1; j > 0; j >>= 1) {
            __syncthreads();
            for (int t = tid; t < SORT_N; t += 256) {
                int ixj = t ^ j;
                if (ixj > t) {
                    float s0 = sScore[t], s1 = sScore[ixj];
                    int   i0 = sIdx[t],   i1 = sIdx[ixj];
                    bool asc = ((t & kk) == 0);
                    // lt: elem[ixj] should precede elem[t] in final order
                    bool lt = (s1 > s0) || (s1 == s0 && i1 < i0);
                    if (lt == asc) {
                        sScore[t] = s1; sScore[ixj] = s0;
                        sIdx[t]   = i1; sIdx[ixj]   = i0;
                    }
                }
            }
        }
    }
    __syncthreads();
    int M = sValid;   // valid candidates occupy sorted slots [0, M)

    // gfx1250 async gather: box of each sorted valid slot, global -> LDS.
    {
        uint64_t sbase = (uint64_t)(uintptr_t)boxBase;
        for (int t = tid; t < SORT_N; t += 256) {
            sSupp[t] = 0;
            if (t < M) {
                unsigned voff = (unsigned)sIdx[t] * 16u;
                unsigned lds  = (unsigned)(uintptr_t)(&sBox[t * 4]);
                asm volatile("global_load_async_to_lds_b128 %0, %1, %2"
                             :: "v"(lds), "v"(voff), "s"(sbase) : "memory");
            }
        }
        asm volatile("s_wait_asynccnt 0" ::: "memory");
    }
    __syncthreads();

    // Greedy NMS: slot i's status is final once all iterations < i complete.
    for (int i = 0; i < M; ++i) {
        if (sSupp[i] == 0) {
            float ax1 = sBox[i * 4 + 0], ay1 = sBox[i * 4 + 1];
            float ax2 = sBox[i * 4 + 2], ay2 = sBox[i * 4 + 3];
            float areaA = (ax2 - ax1 + 1.0f) * (ay2 - ay1 + 1.0f);
            for (int jj = i + 1 + tid; jj < M; jj += 256) {
                if (!sSupp[jj]) {
                    float bx1 = sBox[jj * 4 + 0], by1 = sBox[jj * 4 + 1];
                    float bx2 = sBox[jj * 4 + 2], by2 = sBox[jj * 4 + 3];
                    float ix1 = fmaxf(ax1, bx1), iy1 = fmaxf(ay1, by1);
                    float ix2 = fminf(ax2, bx2), iy2 = fminf(ay2, by2);
                    float iw = fmaxf(ix2 - ix1 + 1.0f, 0.0f);
                    float ih = fmaxf(iy2 - iy1 + 1.0f, 0.0f);
                    float inter = iw * ih;
                    float areaB = (bx2 - bx1 + 1.0f) * (by2 - by1 + 1.0f);
                    float iou = inter / (areaA + areaB - inter);
                    if (iou > NMS_T) sSupp[jj] = 1;
                }
            }
        }
        __syncthreads();
    }

    // Compact kept entries (order irrelevant for rank counting).
    for (int i = tid; i < M; i += 256) {
        if (!sSupp[i]) {
            int slot = atomicAdd(&sCnt, 1);
            keptScore[(size_t)pIdx * SORT_N + slot] = sScore[i];
            keptFlat [(size_t)pIdx * SORT_N + slot] = c * N_ + i; // flat tie key
        }
    }
    __syncthreads();

    if (tid == 0) {
        keptCount[pIdx] = sCnt;
        bool any = (M > 0);          // slot 0 can never be suppressed
        cbScore[pIdx] = any ? sScore[0] : NEG_INF;
        cbFlat[pIdx]  = c * N_;      // sorted position 0
        for (int q = 0; q < 4; ++q)
            cbBox[pIdx * 4 + q] = any ? sBox[q] : 0.0f;
        if (c == 0) {
            // image-empty fallback: class-0 sorted-slot-0 box (topk_b[0])
            int o = sIdx[0];
            for (int q = 0; q < 4; ++q)
                fallback[(path * B_ + b) * 4 + q] = boxBase[o * 4 + q];
        }
    }
}

// ---------------------------------------------------------------------------
// Kernel 3: per (path, image): strict-outrank counts of each class-best vs
// all kept entries -> exact top-50 membership; emit bestS/bestB per class.
// ---------------------------------------------------------------------------
__global__ __launch_bounds__(256) void rank_kernel(
    const float* __restrict__ keptScore, const int* __restrict__ keptFlat,
    const int* __restrict__ keptCount,
    const float* __restrict__ cbScore, const int* __restrict__ cbFlat,
    const float* __restrict__ cbBox, const float* __restrict__ fallback,
    float* __restrict__ bestS, float* __restrict__ bestB)
{
    __shared__ float shS[FG];
    __shared__ int   shF[FG];
    __shared__ int   shC[FG];
    __shared__ int   shR[FG];

    int tid  = threadIdx.x;
    int b    = blockIdx.x;
    int path = blockIdx.y;
    int pB   = path * B_ + b;

    if (tid < FG) {
        shS[tid] = cbScore[pB * FG + tid];
        shF[tid] = cbFlat [pB * FG + tid];
        shC[tid] = keptCount[pB * FG + tid];
        shR[tid] = 0;
    }
    __syncthreads();

    int r[FG];
#pragma unroll
    for (int q = 0; q < FG; ++q) r[q] = 0;

    for (int c = 0; c < FG; ++c) {
        int cnt = shC[c];
        size_t base = (size_t)(pB * FG + c) * SORT_N;
        for (int s = tid; s < cnt; s += 256) {
            float sc = keptScore[base + s];
            int   fl = keptFlat [base + s];
#pragma unroll
            for (int q = 0; q < FG; ++q) {
                if ((sc > shS[q]) || (sc == shS[q] && fl < shF[q])) r[q]++;
            }
        }
    }
#pragma unroll
    for (int q = 0; q < FG; ++q)
        if (r[q]) atomicAdd(&shR[q], r[q]);
    __syncthreads();

    if (tid == 0) {
        float gb[4];
        bool any = false;
        for (int cc = 0; cc < FG && !any; ++cc) {
            if (shC[cc] > 0 && shR[cc] == 0) {
                for (int q = 0; q < 4; ++q) gb[q] = cbBox[(pB * FG + cc) * 4 + q];
                any = true;
            }
        }
        if (!any) {
            for (int q = 0; q < 4; ++q) gb[q] = fallback[pB * 4 + q];
        }
        for (int cc = 0; cc < FG; ++cc) {
            bool present = (shC[cc] > 0) && (shR[cc] < MAXPI);
            bestS[pB * FG + cc] = present ? shS[cc] : NEG_INF;
            for (int q = 0; q < 4; ++q)
                bestB[(pB * FG + cc) * 4 + q] =
                    present ? cbBox[(pB * FG + cc) * 4 + q] : gb[q];
        }
    }
}

// ---------------------------------------------------------------------------
// Kernel 4: IEEE joint product (preserves (-inf)*(-inf)=+inf) + first-max
// argmax; write final boxes.
// ---------------------------------------------------------------------------
__global__ void final_kernel(const float* __restrict__ bestS,
                             const float* __restrict__ bestB,
                             float* __restrict__ out)
{
    int b = threadIdx.x;
    if (b < B_) {
        const float* sA = bestS + (0 * B_ + b) * FG;
        const float* sB = bestS + (1 * B_ + b) * FG;
        float best = sA[0] * sB[0];
        int cls = 0;
        for (int c2 = 1; c2 < FG; ++c2) {
            float j = sA[c2] * sB[c2];
            if (j > best) { best = j; cls = c2; }
        }
        for (int q = 0; q < 4; ++q) {
            out[b * 4 + q]          = bestB[((0 * B_ + b) * FG + cls) * 4 + q];
            out[B_ * 4 + b * 4 + q] = bestB[((1 * B_ + b) * FG + cls) * 4 + q];
        }
    }
}

// ---------------------------------------------------------------------------
extern "C" void kernel_launch(void* const* d_in, const int* in_sizes, int n_in,
                              void* d_out, int out_size, void* d_ws, size_t ws_size,
                              hipStream_t stream) {
    (void)in_sizes; (void)n_in; (void)out_size; (void)ws_size;
    const float* roisA = (const float*)d_in[0];
    const float* clsA  = (const float*)d_in[1];
    const float* bbA   = (const float*)d_in[2];
    const float* infoA = (const float*)d_in[3];
    const float* roisB = (const float*)d_in[4];
    const float* clsB  = (const float*)d_in[5];
    const float* bbB   = (const float*)d_in[6];
    const float* infoB = (const float*)d_in[7];

    char* ws = (char*)d_ws;
    size_t off = 0;
    auto carve = [&](size_t bytes) -> char* {
        char* p = ws + off;
        off += (bytes + 255) / 256 * 256;
        return p;
    };
    float* decBoxes  = (float*)carve(sizeof(float) * 2 * B_ * FG * N_ * 4);
    float* decScores = (float*)carve(sizeof(float) * 2 * B_ * FG * N_);
    float* keptScore = (float*)carve(sizeof(float) * 2 * B_ * FG * SORT_N);
    int*   keptFlat  = (int*)  carve(sizeof(int)   * 2 * B_ * FG * SORT_N);
    int*   keptCount = (int*)  carve(sizeof(int)   * 2 * B_ * FG);
    float* cbScore   = (float*)carve(sizeof(float) * 2 * B_ * FG);
    int*   cbFlat    = (int*)  carve(sizeof(int)   * 2 * B_ * FG);
    float* cbBox     = (float*)carve(sizeof(float) * 2 * B_ * FG * 4);
    float* fallback  = (float*)carve(sizeof(float) * 2 * B_ * 4);
    float* bestS     = (float*)carve(sizeof(float) * 2 * B_ * FG);
    float* bestB     = (float*)carve(sizeof(float) * 2 * B_ * FG * 4);

    int total  = B_ * N_ * FG;
    int blocks = (total + 255) / 256;
    decode_kernel<<<blocks, 256, 0, stream>>>(roisA, bbA, clsA, infoA,
                                              decBoxes, decScores, 0);
    decode_kernel<<<blocks, 256, 0, stream>>>(roisB, bbB, clsB, infoB,
                                              decBoxes, decScores, 1);
    nms_kernel<<<dim3(FG, B_, 2), 256, 0, stream>>>(
        decBoxes, decScores, keptScore, keptFlat, keptCount,
        cbScore, cbFlat, cbBox, fallback);
    rank_kernel<<<dim3(B_, 2), 256, 0, stream>>>(
        keptScore, keptFlat, keptCount, cbScore, cbFlat, cbBox, fallback,
        bestS, bestB);
    final_kernel<<<1, 32, 0, stream>>>(bestS, bestB, (float*)d_out);
}